// ClassifierGCN_23794118820245
// MI455X (gfx1250) — compile-verified
//
#include <hip/hip_runtime.h>
#include <math.h>

#define N_NEURONS 80
#define TBL 256
#define LATENT 128
#define N_GRAPHS 64
#define N_NODES (N_GRAPHS * N_NEURONS)   // 5120
#define N_EDGES (N_NODES * 32)           // 163840
#define N2 (N_NEURONS * N_NEURONS)       // 6400
#define H1 (2 * N2)                      // 12800

typedef float v2f __attribute__((ext_vector_type(2)));
typedef float v8f __attribute__((ext_vector_type(8)));
typedef unsigned int v4u __attribute__((ext_vector_type(4)));
typedef int v4i __attribute__((ext_vector_type(4)));
typedef int v8i __attribute__((ext_vector_type(8)));

__device__ __forceinline__ v8f wmma4(v2f a, v2f b, v8f c) {
  // D = A(16x4,f32) * B(4x16,f32) + C(16x16,f32)
  return __builtin_amdgcn_wmma_f32_16x16x4_f32(false, a, false, b, (short)0, c,
                                               false, false);
}

__device__ __forceinline__ v8f vzero8() {
  v8f z = {0.f, 0.f, 0.f, 0.f, 0.f, 0.f, 0.f, 0.f};
  return z;
}

// ---- Tensor Data Mover: 2D tile (f32) global -> LDS, optional LDS padding ----
// D# per CDNA5 ISA ch.8: group0 {count, lds_addr, global_addr, type=2},
// group1 {data_size, pad ctrl, tensor dims, tile dims, row stride}, g2/g3 = 0.
// This toolchain's builtin is the 6-arg form: (v4u, v8i, v4i, v4i, v8i, i32).
__device__ __forceinline__ void tdm_load_2d_f32(
    unsigned lds_byte_addr, const float* gaddr, unsigned tile_x, unsigned tile_y,
    unsigned long row_stride_elems, unsigned pad_interval_code,
    unsigned pad_amount_code, bool pad_en) {
  unsigned long ga = (unsigned long)(const void*)gaddr;
  v4u g0;
  g0[0] = 1u;                                            // count=1 (user D#)
  g0[1] = lds_byte_addr;                                 // LDS byte address
  g0[2] = (unsigned)(ga & 0xffffffffu);                  // global_addr[31:0]
  g0[3] = (unsigned)((ga >> 32) & 0x01ffffffu) | (2u << 30);  // [56:32]|type=2

  unsigned dw0 = (2u << 16);                             // data_size = 4 bytes
  if (pad_en)
    dw0 |= (1u << 20) | (pad_interval_code << 22) | (pad_amount_code << 25);
  v8i g1;
  g1[0] = (int)dw0;
  g1[1] = (int)((tile_x & 0xffffu) << 16);               // tensor_dim0 lo
  g1[2] = (int)((tile_x >> 16) | ((tile_y & 0xffffu) << 16));  // td0 hi|td1 lo
  g1[3] = (int)((tile_y >> 16) | (tile_x << 16));        // td1 hi | tile_dim0
  g1[4] = (int)(tile_y & 0xffffu);                       // tile_dim1 (dim2=0)
  g1[5] = (int)(row_stride_elems & 0xffffffffu);         // dim0_stride lo
  g1[6] = (int)((row_stride_elems >> 32) & 0xffffu);     // dim0_stride hi
  g1[7] = 0;
  v4i g2 = {0, 0, 0, 0};
  v4i g3 = {0, 0, 0, 0};
  v8i gx = {0, 0, 0, 0, 0, 0, 0, 0};  // extra group (6-arg form), zero-filled
  __builtin_amdgcn_tensor_load_to_lds(g0, g1, g2, g3, gx, 0);
}

// ---------------------------------------------------------------- utilities
__global__ void zero_f32(float* __restrict__ p, int n) {
  int i = blockIdx.x * blockDim.x + threadIdx.x;
  if (i < n) p[i] = 0.0f;
}

// ------------------------------------------------ h = x @ Wg  [5120,256]x[256,128]
__global__ void gcn_xw_kernel(const float* __restrict__ x,
                              const float* __restrict__ Wg,
                              float* __restrict__ h) {
  const int wave = threadIdx.x >> 5;
  const int lane = threadIdx.x & 31;
  const int m = lane & 15;
  const int kk = (lane >> 4) << 1;
  const int rbase = (lane >> 4) * 8;
  const int mBase = blockIdx.x * 16;
  const int nBase = wave * 16;

  v8f acc = vzero8();
  const float* arow = x + (size_t)(mBase + m) * TBL;
  for (int k = 0; k < TBL; k += 4) {
    v2f a, b;
    a.x = arow[k + kk];
    a.y = arow[k + kk + 1];
    b.x = Wg[(k + kk) * LATENT + nBase + m];
    b.y = Wg[(k + kk + 1) * LATENT + nBase + m];
    acc = wmma4(a, b, acc);
  }
#pragma unroll
  for (int r = 0; r < 8; ++r)
    h[(size_t)(mBase + rbase + r) * LATENT + nBase + m] = acc[r];
}

// ---------------------------------------------------------------- degree / norm
__global__ void degree_kernel(const int* __restrict__ dst,
                              float* __restrict__ deg) {
  int e = blockIdx.x * blockDim.x + threadIdx.x;
  if (e < N_EDGES) atomicAdd(&deg[dst[e]], 1.0f);
}

__global__ void dinv_kernel(const float* __restrict__ deg,
                            float* __restrict__ dinv) {
  int i = blockIdx.x * blockDim.x + threadIdx.x;
  if (i < N_NODES) {
    float d = deg[i];
    dinv[i] = (d > 0.0f) ? rsqrtf(fmaxf(d, 1.0f)) : 0.0f;
  }
}

// ------------------------------------- agg[dst] += h[src] * dinv[src]*dinv[dst]
__global__ void scatter_kernel(const int* __restrict__ src,
                               const int* __restrict__ dst,
                               const float* __restrict__ h,
                               const float* __restrict__ dinv,
                               float* __restrict__ agg) {
  int t = blockIdx.x * blockDim.x + threadIdx.x;
  int e = t >> 5;
  int lane = t & 31;
  if (e >= N_EDGES) return;
  int s = src[e], d = dst[e];
  float norm = dinv[s] * dinv[d];
  const float4 v = *(const float4*)(h + (size_t)s * LATENT + lane * 4);
  float* out = agg + (size_t)d * LATENT + lane * 4;
  atomicAdd(out + 0, v.x * norm);
  atomicAdd(out + 1, v.y * norm);
  atomicAdd(out + 2, v.z * norm);
  atomicAdd(out + 3, v.w * norm);
}

// ----------------------- per-graph: z = relu(agg+bg); G = z @ z^T
#define ZSTRIDE 132
__global__ void outer_kernel(const float* __restrict__ agg,
                             const float* __restrict__ bg,
                             float* __restrict__ G) {
  __shared__ float z[N_NEURONS * ZSTRIDE];
  const int g = blockIdx.x;

  for (int i = threadIdx.x; i < N_NEURONS * LATENT; i += blockDim.x) {
    int row = i >> 7;
    int col = i & 127;
    float v = agg[(size_t)(g * N_NEURONS + row) * LATENT + col] + bg[col];
    z[row * ZSTRIDE + col] = fmaxf(v, 0.0f);
  }
  __syncthreads();

  const int wave = threadIdx.x >> 5;
  const int lane = threadIdx.x & 31;
  const int m = lane & 15;
  const int kk = (lane >> 4) << 1;
  const int rbase = (lane >> 4) * 8;

  for (int t = wave; t < 25; t += 8) {
    int mt = t / 5, nt = t % 5;
    const float* za = z + (mt * 16 + m) * ZSTRIDE;
    const float* zb = z + (nt * 16 + m) * ZSTRIDE;
    v8f acc = vzero8();
    for (int k = 0; k < LATENT; k += 4) {
      v2f a, b;
      a.x = za[k + kk];
      a.y = za[k + kk + 1];
      b.x = zb[k + kk];
      b.y = zb[k + kk + 1];
      acc = wmma4(a, b, acc);
    }
    float* out = G + (size_t)g * N2 + nt * 16 + m;
#pragma unroll
    for (int r = 0; r < 8; ++r)
      out[(mt * 16 + rbase + r) * N_NEURONS] = acc[r];
  }
}

// ---- out = act(A[64,K] @ W[K,N] + bias): TDM double-buffered, 16 waves/block
// A tile 64x16 -> LDS stride 17 via TDM pad (1 dw / 16 dw); W tile 16x64 ->
// LDS stride 66 via TDM pad (2 dw / 64 dw). Wave 0 drives the DMA engine.
template <int KDIM, int NDIM, bool SIGMOID>
__global__ __launch_bounds__(512) void gemm64_tdm_kernel(
    const float* __restrict__ A, const float* __restrict__ W,
    const float* __restrict__ bias, float* __restrict__ out) {
  __shared__ float As[2][64 * 17];
  __shared__ float Bs[2][16 * 66];

  const int tid = threadIdx.x;
  const int wave = tid >> 5;
  const int lane = tid & 31;
  const int mi = wave >> 2;
  const int ni = wave & 3;
  const int m = lane & 15;
  const int kk = (lane >> 4) << 1;
  const int rbase = (lane >> 4) * 8;
  const int nBase = blockIdx.x * 64;
  const int NCHUNK = KDIM / 16;

  if (wave == 0) {  // prologue: stream chunk 0 into buffer 0
    tdm_load_2d_f32((unsigned)(size_t)(void*)&As[0][0], A, 16, 64,
                    (unsigned long)KDIM, /*interval 16dw*/ 3, /*pad 1dw*/ 0, true);
    tdm_load_2d_f32((unsigned)(size_t)(void*)&Bs[0][0], W + (size_t)0 * NDIM + nBase,
                    64, 16, (unsigned long)NDIM, /*interval 64dw*/ 5, /*pad 2dw*/ 1,
                    true);
  }

  v8f acc = vzero8();

  for (int i = 0; i < NCHUNK; ++i) {
    const int cur = i & 1;
    if (wave == 0) {
      if (i + 1 < NCHUNK) {  // stream chunk i+1 into the other buffer
        const int nxt = (i + 1) & 1;
        const int k0n = (i + 1) * 16;
        tdm_load_2d_f32((unsigned)(size_t)(void*)&As[nxt][0], A + k0n, 16, 64,
                        (unsigned long)KDIM, 3, 0, true);
        tdm_load_2d_f32((unsigned)(size_t)(void*)&Bs[nxt][0],
                        W + (size_t)k0n * NDIM + nBase, 64, 16,
                        (unsigned long)NDIM, 5, 1, true);
        __builtin_amdgcn_s_wait_tensorcnt(2);  // chunk i's 2 DMAs done (in-order)
      } else {
        __builtin_amdgcn_s_wait_tensorcnt(0);
      }
    }
    __syncthreads();  // buffer `cur` visible to all 16 waves

    const float* ap = &As[cur][(mi * 16 + m) * 17];
    const float* bp = &Bs[cur][ni * 16 + m];
#pragma unroll
    for (int ks = 0; ks < 16; ks += 4) {
      v2f a, b;
      a.x = ap[ks + kk];
      a.y = ap[ks + kk + 1];
      b.x = bp[(ks + kk) * 66];
      b.y = bp[(ks + kk + 1) * 66];
      acc = wmma4(a, b, acc);
    }
    __syncthreads();  // all waves done with `cur` before it is re-filled
  }

  const int ncol = nBase + ni * 16 + m;
  const float bv = bias[ncol];
#pragma unroll
  for (int r = 0; r < 8; ++r) {
    float v = acc[r] + bv;
    if (SIGMOID)
      v = 1.0f / (1.0f + expf(-v));
    else
      v = fmaxf(v, 0.0f);
    out[(size_t)(mi * 16 + rbase + r) * NDIM + ncol] = v;
  }
}

// ---------------------------------------------------------------------------
extern "C" void kernel_launch(void* const* d_in, const int* in_sizes, int n_in,
                              void* d_out, int out_size, void* d_ws,
                              size_t ws_size, hipStream_t stream) {
  const float* x  = (const float*)d_in[0];
  const int*   ei = (const int*)d_in[1];
  const float* Wg = (const float*)d_in[2];
  const float* bg = (const float*)d_in[3];
  const float* W1 = (const float*)d_in[4];
  const float* b1 = (const float*)d_in[5];
  const float* W2 = (const float*)d_in[6];
  const float* b2 = (const float*)d_in[7];
  float* out = (float*)d_out;
  float* ws  = (float*)d_ws;

  float* h    = ws;                      // 5120*128
  float* agg  = h + N_NODES * LATENT;    // 5120*128
  float* deg  = agg + N_NODES * LATENT;  // 5120
  float* dinv = deg + N_NODES;           // 5120
  float* G    = dinv + N_NODES;          // 64*6400
  float* z2   = G + N_GRAPHS * N2;       // 64*12800

  const int* src = ei;
  const int* dst = ei + N_EDGES;

  const int nz = N_NODES * LATENT + N_NODES;
  zero_f32<<<(nz + 255) / 256, 256, 0, stream>>>(agg, nz);

  gcn_xw_kernel<<<N_NODES / 16, 256, 0, stream>>>(x, Wg, h);
  degree_kernel<<<(N_EDGES + 255) / 256, 256, 0, stream>>>(dst, deg);
  dinv_kernel<<<(N_NODES + 255) / 256, 256, 0, stream>>>(deg, dinv);
  scatter_kernel<<<(N_EDGES * 32) / 256, 256, 0, stream>>>(src, dst, h, dinv,
                                                           agg);
  outer_kernel<<<N_GRAPHS, 256, 0, stream>>>(agg, bg, G);

  gemm64_tdm_kernel<N2, H1, false><<<H1 / 64, 512, 0, stream>>>(G, W1, b1, z2);
  gemm64_tdm_kernel<H1, N2, true><<<N2 / 64, 512, 0, stream>>>(z2, W2, b2, out);
}